// Weak_Supv_Loss_21354577395725
// MI455X (gfx1250) — compile-verified
//
#include <hip/hip_runtime.h>
#include <hip/hip_bf16.h>

typedef __attribute__((ext_vector_type(2))) float v2f;
typedef __attribute__((ext_vector_type(8))) float v8f;

#define NB 16
#define NC 3
#define NR 512
#define NA 512
#define NG 4096
#define NK 4
#define R_TOP 20
#define NS 41          // number of shifts (-20..+20)
#define PREN 52        // prefix-boundary entries 0..51
#define SMALL_PROB 0.3f

// One block per detection-group g; 4 wave32s, wave k handles detection slot k.
__global__ __launch_bounds__(128) void wsl_kernel(
    const float* __restrict__ pred,
    const int*   __restrict__ det_b,
    const int*   __restrict__ det_c,
    const int*   __restrict__ det_r,
    const int*   __restrict__ det_a,
    const int*   __restrict__ radus_r,
    float*       __restrict__ out)
{
    const int g    = blockIdx.x;
    const int tid  = threadIdx.x;
    const int lane = tid & 31;
    const int k    = tid >> 5;          // wave id == detection slot

    __shared__ float rs_sel[NK][PREN];  // banded row sums, selected channel
    __shared__ float rs_all[NK][PREN];  // banded row sums, all channels
    __shared__ float pre_sel[NK][PREN]; // exclusive prefix sums
    __shared__ float pre_all[NK][PREN];
    __shared__ float val3buf[NK][48];   // per-(k,shift) val3, padded to 48 cols
    __shared__ float sebuf[48];         // se_sum per shift

    const int gi  = g * NK + k;
    const int b   = det_b[gi];
    const int cs  = det_c[gi];
    const int dr  = det_r[gi];
    const int da  = det_a[gi];
    const int rad = radus_r[gi];

    // ---- Step 1: banded row sums over the 5-wide column window around da.
    // Rows needed: dr-25 .. dr+25 (j = 0..50); out-of-range rows contribute 0,
    // which makes the reference's SAT clipping exact.
    int a0 = da - 2; if (a0 < 0) a0 = 0;
    int a1 = da + 2; if (a1 > NA - 1) a1 = NA - 1;
    for (int j = lane; j < PREN; j += 32) {
        float accSel = 0.f, accAll = 0.f;
        const int row = dr - 25 + j;
        if (row >= 0 && row < NR) {
            #pragma unroll
            for (int c = 0; c < NC; ++c) {
                const float* rp = pred + (((size_t)(b * NC + c) * NR + row) * NA);
                float s = 0.f;
                for (int a = a0; a <= a1; ++a) s += rp[a];
                accAll += s;
                if (c == cs) accSel = s;
            }
        }
        rs_sel[k][j] = accSel;
        rs_all[k][j] = accAll;
    }
    __syncthreads();

    // ---- Step 2: serial exclusive prefix (51 useful rows; cheap).
    if (lane == 0) {
        float acc = 0.f;
        for (int i = 0; i < PREN; ++i) { pre_sel[k][i] = acc; acc += rs_sel[k][i]; }
    } else if (lane == 1) {
        float acc = 0.f;
        for (int i = 0; i < PREN; ++i) { pre_all[k][i] = acc; acc += rs_all[k][i]; }
    }
    __syncthreads();

    // ---- Step 3: per-shift window sums.
    // i1 = shift - rad + 25 in [0,45];  i2 = shift + rad + 26 in [6,51].
    for (int s = lane; s < 48; s += 32) {
        float v3 = 0.f;
        if (s < NS) {
            const int shift = s - R_TOP;
            const int i1 = shift - rad + 25;
            const int i2 = shift + rad + 26;
            const float w = expf(-fabsf((float)shift) / 200.f);
            const float val1 = (pre_sel[k][i2] - pre_sel[k][i1]) * w;
            const float val2 = (pre_all[k][i2] - pre_all[k][i1]) * w;
            v3 = 0.5f * val1 + 0.5f * (val2 - val1);   // RT_SUM = 0.5
        }
        val3buf[k][s] = v3;
    }
    __syncthreads();

    // ---- Step 4: K-reduction via tensor core (wave 0 only; wave-uniform branch,
    // so EXEC is all-ones inside as WMMA requires).
    // D = A(16x4, all ones) * B(4x16, B[k][n] = val3[k][shifttile n]) + 0
    //  => every row of D holds the column sums  se[n] = sum_k val3[k][n].
    if (k == 0) {
        v2f amat; amat[0] = 1.0f; amat[1] = 1.0f;      // all-ones A matrix
        const int klo = (lane < 16) ? 0 : 2;           // B: VGPR0 = rows K0/K2, VGPR1 = K1/K3
        #pragma unroll
        for (int t = 0; t < 3; ++t) {
            const int col = t * 16 + (lane & 15);
            v2f bmat;
            bmat[0] = val3buf[klo][col];
            bmat[1] = val3buf[klo + 1][col];
            v8f cacc = {};
            v8f d = __builtin_amdgcn_wmma_f32_16x16x4_f32(
                /*neg_a=*/false, amat, /*neg_b=*/false, bmat,
                /*c_mod=*/(short)0, cacc, /*reuse_a=*/false, /*reuse_b=*/false);
            if (lane < 16 && col < NS) {               // M=0 row lives in d[0], lanes 0..15
                const float se = d[0];
                sebuf[col] = se;
                out[(size_t)g * NS + col] = se;
            }
        }
    }
    __syncthreads();

    // ---- Step 5: argmax (first occurrence), prb_conf, prb_max.
    if (tid == 0) {
        float best = sebuf[0];
        int m = 0;
        for (int s = 1; s < NS; ++s) {
            const float v = sebuf[s];
            if (v > best) { best = v; m = s; }         // strict > keeps first index
        }
        const int ind = m - R_TOP;
        float pmax = SMALL_PROB;
        for (int kk = 0; kk < NK; ++kk) {
            const int i  = g * NK + kk;
            const int bb = det_b[i];
            const int aa = det_a[i];
            int rr = det_r[i] + ind;
            rr = rr < 0 ? 0 : (rr > NR - 1 ? NR - 1 : rr);
            #pragma unroll
            for (int c = 0; c < NC; ++c) {
                const float v = pred[(((size_t)(bb * NC + c) * NR + rr) * NA) + aa];
                pmax = fmaxf(pmax, v);
            }
        }
        out[(size_t)NG * NS + g]            = (float)ind;  // ind_shift
        out[(size_t)NG * NS + NG + g]       = best;        // prb_conf
        out[(size_t)NG * NS + 2 * NG + g]   = pmax;        // prb_max
    }
}

extern "C" void kernel_launch(void* const* d_in, const int* in_sizes, int n_in,
                              void* d_out, int out_size, void* d_ws, size_t ws_size,
                              hipStream_t stream) {
    const float* pred   = (const float*)d_in[0];
    const int*   det_b  = (const int*)d_in[1];
    const int*   det_c  = (const int*)d_in[2];
    const int*   det_r  = (const int*)d_in[3];
    const int*   det_a  = (const int*)d_in[4];
    const int*   radusr = (const int*)d_in[5];
    float*       out    = (float*)d_out;

    wsl_kernel<<<dim3(NG), dim3(128), 0, stream>>>(
        pred, det_b, det_c, det_r, det_a, radusr, out);
}